// Embedding_50251117363824
// MI455X (gfx1250) — compile-verified
//
#include <hip/hip_runtime.h>

// Embedding gather for MI455X (gfx1250, wave32).
// One wave32 = one (b,l) position; lane owns 4 of the 128 dims (float4).
// Index chase is wave-uniform -> scalarized via readfirstlane (s_load + s_cbranch).
// Output is a 128MB stream -> non-temporal stores keep the ~28MB of tables in L2.

typedef __attribute__((ext_vector_type(4))) float v4f;

__global__ __launch_bounds__(256) void embed_gather_kernel(
    const int*   __restrict__ feature_ids,        // [B*L]
    const float* __restrict__ feature_values,     // [B*L]
    const float* __restrict__ cat_table,          // [N_CAT, 128]
    const float* __restrict__ num_weight,         // [N_NUM, 128]
    const float* __restrict__ num_bias,           // [N_NUM, 128]
    const int*   __restrict__ input_to_numeric,   // [VOCAB+1]
    const int*   __restrict__ input_to_categorical,// [VOCAB+1]
    float*       __restrict__ out,                // [B*L, 128]
    int n_pos)
{
    const int lane = threadIdx.x & 31;
    // wave index within grid; uniform across the wave -> force into SGPR
    const int pos = __builtin_amdgcn_readfirstlane(
        (int)(blockIdx.x * 8 + (threadIdx.x >> 5)));
    if (pos >= n_pos) return;

    // id is one value per wave: scalar load path
    const int id = __builtin_amdgcn_readfirstlane(feature_ids[pos]);
    const int n_idx = input_to_numeric[id];   // SGPR addr -> s_load
    const int d = lane << 2;                  // this lane's dim offset (0..124)

    v4f r;
    if (n_idx > 0) {
        // numerical path: out = w[row] * value + b[row]
        const int vi = __builtin_amdgcn_readfirstlane(
            __builtin_bit_cast(int, feature_values[pos]));
        const float v = __builtin_bit_cast(float, vi);
        const size_t base = (size_t)(n_idx - 1) * 128 + d;
        const v4f w = *(const v4f*)(num_weight + base);
        const v4f b = *(const v4f*)(num_bias + base);
        r.x = fmaf(w.x, v, b.x);
        r.y = fmaf(w.y, v, b.y);
        r.z = fmaf(w.z, v, b.z);
        r.w = fmaf(w.w, v, b.w);
    } else {
        // categorical path: pure row gather (row 0 = zero padding row)
        const int c_idx = input_to_categorical[id];
        r = *(const v4f*)(cat_table + (size_t)c_idx * 128 + d);
    }

    // streaming output: non-temporal so the tables stay resident in L2
    __builtin_nontemporal_store(r, (v4f*)(out + (size_t)pos * 128 + d));
}

extern "C" void kernel_launch(void* const* d_in, const int* in_sizes, int n_in,
                              void* d_out, int out_size, void* d_ws, size_t ws_size,
                              hipStream_t stream) {
    const int*   feature_ids          = (const int*)  d_in[0];
    const float* feature_values       = (const float*)d_in[1];
    const float* cat_table            = (const float*)d_in[2];
    const float* num_weight           = (const float*)d_in[3];
    const float* num_bias             = (const float*)d_in[4];
    const int*   input_to_numeric     = (const int*)  d_in[5];
    const int*   input_to_categorical = (const int*)  d_in[6];
    float*       out                  = (float*)d_out;

    const int n_pos = in_sizes[0];              // B*L = 262144
    const int waves_per_block = 8;              // 256 threads / wave32
    const int blocks = (n_pos + waves_per_block - 1) / waves_per_block;

    embed_gather_kernel<<<blocks, 256, 0, stream>>>(
        feature_ids, feature_values, cat_table, num_weight, num_bias,
        input_to_numeric, input_to_categorical, out, n_pos);
}